// VisionMambaSingle_59923383714045
// MI455X (gfx1250) — compile-verified
//
#include <hip/hip_runtime.h>
#include <hip/hip_bf16.h>
#include <math.h>
#include <stdint.h>

#define D_MODEL 192
#define D_INNER 384
#define D_STATE 16
#define DT_RANK 12
#define K_CONV  4
#define DEPTH   24
#define BATCH   8
#define L_SEQ   196
#define M_ROWS  (BATCH * L_SEQ)   // 1568, multiple of 16
#define E_PROJ  (2 * D_INNER)     // 768
#define DBL_N   (DT_RANK + 2 * D_STATE) // 44
#define LN_EPS  1e-5f

typedef float v2f __attribute__((ext_vector_type(2)));
typedef float v8f __attribute__((ext_vector_type(8)));

#define WMMA_F32(A_, B_, C_) \
  __builtin_amdgcn_wmma_f32_16x16x4_f32(false, (A_), false, (B_), (short)0, (C_), false, false)

#if defined(__AMDGCN__) && __has_builtin(__builtin_amdgcn_global_load_async_to_lds_b128)
#define USE_ASYNC_LDS 1
#else
#define USE_ASYNC_LDS 0
#endif

#if USE_ASYNC_LDS
typedef int v4i __attribute__((vector_size(4 * sizeof(int))));
typedef __attribute__((address_space(1))) v4i* as1_v4i;
typedef __attribute__((address_space(3))) v4i* as3_v4i;

__device__ __forceinline__ void async_b128(const float* g, const float* l) {
  __builtin_amdgcn_global_load_async_to_lds_b128(
      (as1_v4i)(uintptr_t)g, (as3_v4i)(uint32_t)(uintptr_t)l, 0, 0);
}
__device__ __forceinline__ void wait_async(int n) {
#if __has_builtin(__builtin_amdgcn_s_wait_asynccnt)
  if (n == 0) __builtin_amdgcn_s_wait_asynccnt(0);
  else        __builtin_amdgcn_s_wait_asynccnt(10);
#else
  if (n == 0) asm volatile("s_wait_asynccnt 0x0" ::: "memory");
  else        asm volatile("s_wait_asynccnt 0xa" ::: "memory");
#endif
}
#endif

// ---------------------------------------------------------------------------
// fp32 GEMM via V_WMMA_F32_16X16X4_F32, register-blocked 4x in N.
// C[m][n] = sum_k A[m][k] * W[n][k]   (A: MxK row-major, W: NxK row-major)
// One wave computes a 16x64 strip of C.
//
// Data movement (CDNA5 path): each K-chunk of 16 is staged into a per-wave
// LDS slice with GLOBAL_LOAD_ASYNC_TO_LDS_B128 (10 issues/lane/chunk, tracked
// by ASYNCcnt), double-buffered so the async DMA of chunk c+1 overlaps the
// WMMAs of chunk c; fragments are then read with short-latency ds_load_b64.
// Falls back to direct global loads if the async builtin is unavailable.
// Requires N % 64 == 0, M % 16 == 0, K % 16 == 0.
// ---------------------------------------------------------------------------
#define GEMM_WPB 4          // waves per block
#define GEMM_KC  16         // K-chunk (floats)
#define GEMM_TILE_F ((16 + 64) * GEMM_KC)   // A(16xKC) + W(64xKC) floats

__global__ __launch_bounds__(GEMM_WPB * 32)
void gemm_wmma_f32_nb4(const float* __restrict__ A,
                       const float* __restrict__ W,
                       float* __restrict__ C,
                       int M, int N, int K) {
#if USE_ASYNC_LDS
  __shared__ float lds[GEMM_WPB * 2 * GEMM_TILE_F];   // 40 KB
#endif
  int wslot = threadIdx.x >> 5;
  int wave  = blockIdx.x * GEMM_WPB + wslot;
  int tnb   = N >> 6;                       // 64-column blocks
  int total = (M >> 4) * tnb;
  if (wave >= total) return;
  int tm = wave / tnb;
  int tn = wave - tm * tnb;

  int lane = threadIdx.x & 31;
  int half = lane >> 4;                     // 0: K pair {0,1}; 1: K pair {2,3}
  int r    = lane & 15;

  v8f acc0 = {}, acc1 = {}, acc2 = {}, acc3 = {};

#if USE_ASYNC_LDS
  const int lrow = lane >> 2;               // 0..7
  const int kx4  = (lane & 3) * 4;          // 0,4,8,12
  float* slice = &lds[wslot * 2 * GEMM_TILE_F];

  // Issue async copy of K-chunk [kbase, kbase+16) into buffer `buf`.
  auto issue_chunk = [&](int kbase, int buf) {
    float* dstA = slice + buf * GEMM_TILE_F;            // [16][KC]
    float* dstW = dstA + 16 * GEMM_KC;                  // [64][KC]
#pragma unroll
    for (int q = 0; q < 2; ++q) {                       // A: 16 rows
      int row = q * 8 + lrow;
      async_b128(A + (size_t)(tm * 16 + row) * K + kbase + kx4,
                 dstA + row * GEMM_KC + kx4);
    }
#pragma unroll
    for (int q = 0; q < 8; ++q) {                       // W: 64 rows
      int row = q * 8 + lrow;
      async_b128(W + (size_t)(tn * 64 + row) * K + kbase + kx4,
                 dstW + row * GEMM_KC + kx4);
    }
  };

  int nc = K / GEMM_KC;
  issue_chunk(0, 0);
  for (int c = 0; c < nc; ++c) {
    if (c + 1 < nc) {
      issue_chunk((c + 1) * GEMM_KC, (c + 1) & 1);
      wait_async(10);          // chunk c complete; chunk c+1 in flight
    } else {
      wait_async(0);
    }
    const float* Ab = slice + (c & 1) * GEMM_TILE_F;
    const float* Wb = Ab + 16 * GEMM_KC;
#pragma unroll
    for (int ks = 0; ks < GEMM_KC; ks += 4) {
      int kl = ks + half * 2;
      v2f a;  a.x  = Ab[r * GEMM_KC + kl];
              a.y  = Ab[r * GEMM_KC + kl + 1];
      v2f b0; b0.x = Wb[(r)      * GEMM_KC + kl]; b0.y = Wb[(r)      * GEMM_KC + kl + 1];
      v2f b1; b1.x = Wb[(16 + r) * GEMM_KC + kl]; b1.y = Wb[(16 + r) * GEMM_KC + kl + 1];
      v2f b2; b2.x = Wb[(32 + r) * GEMM_KC + kl]; b2.y = Wb[(32 + r) * GEMM_KC + kl + 1];
      v2f b3; b3.x = Wb[(48 + r) * GEMM_KC + kl]; b3.y = Wb[(48 + r) * GEMM_KC + kl + 1];
      acc0 = WMMA_F32(a, b0, acc0);
      acc1 = WMMA_F32(a, b1, acc1);
      acc2 = WMMA_F32(a, b2, acc2);
      acc3 = WMMA_F32(a, b3, acc3);
    }
  }
#else
  // Fallback: direct global loads.
  const float* Arow = A + (size_t)(tm * 16 + r) * K;
  const float* W0   = W + (size_t)(tn * 64 + r) * K;
  const float* W1   = W0 + (size_t)16 * K;
  const float* W2   = W1 + (size_t)16 * K;
  const float* W3   = W2 + (size_t)16 * K;
  for (int k = 0; k < K; k += 4) {
    int kk = k + half * 2;
    v2f a;  a.x  = Arow[kk]; a.y  = Arow[kk + 1];
    v2f b0; b0.x = W0[kk];   b0.y = W0[kk + 1];
    v2f b1; b1.x = W1[kk];   b1.y = W1[kk + 1];
    v2f b2; b2.x = W2[kk];   b2.y = W2[kk + 1];
    v2f b3; b3.x = W3[kk];   b3.y = W3[kk + 1];
    acc0 = WMMA_F32(a, b0, acc0);
    acc1 = WMMA_F32(a, b1, acc1);
    acc2 = WMMA_F32(a, b2, acc2);
    acc3 = WMMA_F32(a, b3, acc3);
  }
#endif

  int m0 = tm * 16 + half * 8;
  int n0 = tn * 64 + r;
#pragma unroll
  for (int i = 0; i < 8; ++i) {
    float* Crow = C + (size_t)(m0 + i) * N + n0;
    Crow[0]  = acc0[i];
    Crow[16] = acc1[i];
    Crow[32] = acc2[i];
    Crow[48] = acc3[i];
  }
}

// ---------------------------------------------------------------------------
// Patch embedding as a gathered-A WMMA GEMM: M=1568, N=192, K=768 (c,i,j).
// hid[b*196+l][n] = sum_{c,i,j} x[b][c][ph*16+i][pw*16+j] * patch_w[n][c][i][j] + pb[n]
// ---------------------------------------------------------------------------
__global__ void patch_embed_wmma(const float* __restrict__ x,
                                 const float* __restrict__ pw,
                                 const float* __restrict__ pb,
                                 float* __restrict__ hid) {
  int wave = blockIdx.x * (blockDim.x >> 5) + (threadIdx.x >> 5);
  const int TN = D_MODEL / 16;  // 12
  int tm = wave / TN, tn = wave - tm * TN;
  if (tm >= (M_ROWS / 16)) return;

  int lane = threadIdx.x & 31;
  int half = lane >> 4;
  int r    = lane & 15;

  int mrow = tm * 16 + r;
  int b    = mrow / L_SEQ;
  int lpos = mrow - b * L_SEQ;
  int ph = lpos / 14, pwid = lpos - ph * 14;
  const float* xb = x + (size_t)b * 3 * 224 * 224;
  const float* Wrow = pw + (size_t)(tn * 16 + r) * 768;

  v8f acc = {};
  for (int k = 0; k < 768; k += 4) {
    int kk = k + half * 2;                   // even => j even, j+1 stays in-row
    int c = kk >> 8, rem = kk & 255;
    int i = rem >> 4, j = rem & 15;
    const float* px = xb + ((size_t)c * 224 + (ph * 16 + i)) * 224 + (pwid * 16 + j);
    v2f a; a.x = px[0]; a.y = px[1];
    v2f bb; bb.x = Wrow[kk]; bb.y = Wrow[kk + 1];
    acc = WMMA_F32(a, bb, acc);
  }

  int m0 = tm * 16 + half * 8;
  int n0 = tn * 16 + r;
  float bias = pb[n0];
#pragma unroll
  for (int i = 0; i < 8; ++i)
    hid[(size_t)(m0 + i) * D_MODEL + n0] = acc[i] + bias;
}

// ---------------------------------------------------------------------------
// res += h; out = layernorm(res) * w + b.  One wave (32 lanes) per row of 192.
// ---------------------------------------------------------------------------
__global__ void add_ln_kernel(const float* __restrict__ h,
                              float* __restrict__ res,
                              const float* __restrict__ w,
                              const float* __restrict__ b,
                              float* __restrict__ out) {
  int row  = blockIdx.x * (blockDim.x >> 5) + (threadIdx.x >> 5);
  int lane = threadIdx.x & 31;
  const float* hr = h + (size_t)row * D_MODEL;
  float* rr = res + (size_t)row * D_MODEL;

  float v[6];
  float s = 0.f;
#pragma unroll
  for (int i = 0; i < 6; ++i) {
    int c = lane + i * 32;
    float t = rr[c] + hr[c];
    rr[c] = t;
    v[i] = t;
    s += t;
  }
#pragma unroll
  for (int m = 16; m; m >>= 1) s += __shfl_xor(s, m, 32);
  float mu = s * (1.f / D_MODEL);
  float s2 = 0.f;
#pragma unroll
  for (int i = 0; i < 6; ++i) { float d = v[i] - mu; s2 += d * d; }
#pragma unroll
  for (int m = 16; m; m >>= 1) s2 += __shfl_xor(s2, m, 32);
  float rstd = rsqrtf(s2 * (1.f / D_MODEL) + LN_EPS);
#pragma unroll
  for (int i = 0; i < 6; ++i) {
    int c = lane + i * 32;
    out[(size_t)row * D_MODEL + c] = (v[i] - mu) * rstd * w[c] + b[c];
  }
}

// ---------------------------------------------------------------------------
// Per sequence row m=(b,l): depthwise causal conv + SiLU -> xc; then
// dbl = x_proj(xc) (44), Bm/Cm split, dt = softplus(dt_proj(dbl[:12]) + dtb).
// One block of 128 threads per row.
// ---------------------------------------------------------------------------
__global__ void conv_xproj_kernel(const float* __restrict__ xz,
                                  const float* __restrict__ cw,
                                  const float* __restrict__ cb,
                                  const float* __restrict__ xpw,
                                  const float* __restrict__ dtw,
                                  const float* __restrict__ dtb,
                                  float* __restrict__ xc,
                                  float* __restrict__ dt,
                                  float* __restrict__ Bm,
                                  float* __restrict__ Cm) {
  __shared__ float xcs[D_INNER];
  __shared__ float dbls[DBL_N];
  int m = blockIdx.x;
  int b = m / L_SEQ, l = m - b * L_SEQ;
  int tid = threadIdx.x;

  for (int d = tid; d < D_INNER; d += blockDim.x) {
    float acc = cb[d];
#pragma unroll
    for (int j = 0; j < K_CONV; ++j) {
      int lq = l - (K_CONV - 1) + j;
      if (lq >= 0)
        acc += xz[(size_t)(b * L_SEQ + lq) * E_PROJ + d] * cw[d * K_CONV + j];
    }
    float sv = acc / (1.f + __expf(-acc));  // SiLU
    xcs[d] = sv;
    xc[(size_t)m * D_INNER + d] = sv;
  }
  __syncthreads();

  int wv = tid >> 5, lane = tid & 31, nwv = blockDim.x >> 5;
  for (int o = wv; o < DBL_N; o += nwv) {
    float p = 0.f;
    for (int e = lane; e < D_INNER; e += 32) p += xcs[e] * xpw[(size_t)o * D_INNER + e];
#pragma unroll
    for (int msk = 16; msk; msk >>= 1) p += __shfl_xor(p, msk, 32);
    if (lane == 0) dbls[o] = p;
  }
  __syncthreads();

  if (tid < D_STATE)
    Bm[(size_t)m * D_STATE + tid] = dbls[DT_RANK + tid];
  else if (tid < 2 * D_STATE)
    Cm[(size_t)m * D_STATE + (tid - D_STATE)] = dbls[DT_RANK + D_STATE + (tid - D_STATE)];

  for (int d = tid; d < D_INNER; d += blockDim.x) {
    float acc = dtb[d];
#pragma unroll
    for (int r = 0; r < DT_RANK; ++r) acc += dbls[r] * dtw[d * DT_RANK + r];
    dt[(size_t)m * D_INNER + d] = (acc > 20.f) ? acc : log1pf(__expf(acc));
  }
}

// ---------------------------------------------------------------------------
// Selective scan + skip + gate.  One lane per (b, d): 16-state recurrence in
// registers over L=196 steps.  y[m][d] = (sum_s h*C + D*xc) * silu(z).
// Uses hardware v_exp_f32 (__expf) — 16 exps/step on the serial critical path.
// ---------------------------------------------------------------------------
__global__ void scan_gate_kernel(const float* __restrict__ xz,
                                 const float* __restrict__ xc,
                                 const float* __restrict__ dt,
                                 const float* __restrict__ Bm,
                                 const float* __restrict__ Cm,
                                 const float* __restrict__ a_log,
                                 const float* __restrict__ dparam,
                                 float* __restrict__ y) {
  int idx = blockIdx.x * blockDim.x + threadIdx.x;
  if (idx >= BATCH * D_INNER) return;
  int b = idx / D_INNER, d = idx - b * D_INNER;

  float A[D_STATE];
#pragma unroll
  for (int s = 0; s < D_STATE; ++s) A[s] = -__expf(a_log[(size_t)d * D_STATE + s]);
  float Dp = dparam[d];
  float h[D_STATE];
#pragma unroll
  for (int s = 0; s < D_STATE; ++s) h[s] = 0.f;

  for (int l = 0; l < L_SEQ; ++l) {
    int m = b * L_SEQ + l;
    float dtv = dt[(size_t)m * D_INNER + d];
    float xv  = xc[(size_t)m * D_INNER + d];
    float dx  = dtv * xv;
    const float* Br = Bm + (size_t)m * D_STATE;
    const float* Cr = Cm + (size_t)m * D_STATE;
    float acc = 0.f;
#pragma unroll
    for (int s = 0; s < D_STATE; ++s) {
      h[s] = __expf(dtv * A[s]) * h[s] + dx * Br[s];
      acc += h[s] * Cr[s];
    }
    acc += Dp * xv;
    float zv = xz[(size_t)m * E_PROJ + D_INNER + d];
    acc *= zv / (1.f + __expf(-zv));   // * silu(z)
    y[(size_t)m * D_INNER + d] = acc;
  }
}

// ---------------------------------------------------------------------------
extern "C" void kernel_launch(void* const* d_in, const int* in_sizes, int n_in,
                              void* d_out, int out_size, void* d_ws, size_t ws_size,
                              hipStream_t stream) {
  const float* x        = (const float*)d_in[0];
  const float* patch_w  = (const float*)d_in[1];
  const float* patch_b  = (const float*)d_in[2];
  const float* norm_w   = (const float*)d_in[3];
  const float* norm_b   = (const float*)d_in[4];
  const float* in_proj  = (const float*)d_in[5];
  const float* conv_w   = (const float*)d_in[6];
  const float* conv_b   = (const float*)d_in[7];
  const float* x_proj   = (const float*)d_in[8];
  const float* dt_projw = (const float*)d_in[9];
  const float* dt_projb = (const float*)d_in[10];
  const float* A_log    = (const float*)d_in[11];
  const float* D_param  = (const float*)d_in[12];
  const float* out_proj = (const float*)d_in[13];
  const float* normf_w  = (const float*)d_in[14];
  const float* normf_b  = (const float*)d_in[15];
  float* out = (float*)d_out;

  const size_t nMD  = (size_t)M_ROWS * D_MODEL;   // 301056
  const size_t nE   = (size_t)M_ROWS * E_PROJ;    // 1204224
  const size_t nDI  = (size_t)M_ROWS * D_INNER;   // 602112
  const size_t nST  = (size_t)M_ROWS * D_STATE;   // 25088
  const size_t need = (3 * nMD + nE + 3 * nDI + 2 * nST) * sizeof(float);
  if (ws_size < need) return;

  float* ws  = (float*)d_ws;
  float* hid = ws;            ws += nMD;
  float* res = ws;            ws += nMD;
  float* ln  = ws;            ws += nMD;
  float* xz  = ws;            ws += nE;
  float* xc  = ws;            ws += nDI;
  float* dtB = ws;            ws += nDI;
  float* yB  = ws;            ws += nDI;
  float* BmB = ws;            ws += nST;
  float* CmB = ws;            ws += nST;

  // Patch embedding: 98*12 = 1176 tiles, 8 waves/block -> 147 blocks.
  patch_embed_wmma<<<147, 256, 0, stream>>>(x, patch_w, patch_b, hid);
  (void)hipMemsetAsync(res, 0, nMD * sizeof(float), stream);

  for (int layer = 0; layer < DEPTH; ++layer) {
    const float* nw  = norm_w   + (size_t)layer * D_MODEL;
    const float* nb  = norm_b   + (size_t)layer * D_MODEL;
    const float* ipw = in_proj  + (size_t)layer * E_PROJ * D_MODEL;
    const float* cw  = conv_w   + (size_t)layer * D_INNER * K_CONV;
    const float* cb  = conv_b   + (size_t)layer * D_INNER;
    const float* xpw = x_proj   + (size_t)layer * DBL_N * D_INNER;
    const float* dtw = dt_projw + (size_t)layer * D_INNER * DT_RANK;
    const float* dtb = dt_projb + (size_t)layer * D_INNER;
    const float* al  = A_log    + (size_t)layer * D_INNER * D_STATE;
    const float* dp  = D_param  + (size_t)layer * D_INNER;
    const float* opw = out_proj + (size_t)layer * D_MODEL * D_INNER;

    // res += h; ln = LN(res).  1568 rows, 8 waves/block -> 196 blocks.
    add_ln_kernel<<<196, 256, 0, stream>>>(hid, res, nw, nb, ln);

    // xz = ln @ ipw^T : 1176 tile-waves, 4 waves/block -> 294 blocks.
    gemm_wmma_f32_nb4<<<294, GEMM_WPB * 32, 0, stream>>>(ln, ipw, xz,
                                                         M_ROWS, E_PROJ, D_MODEL);

    // conv + silu + x_proj + dt_proj: one block per row.
    conv_xproj_kernel<<<M_ROWS, 128, 0, stream>>>(xz, cw, cb, xpw, dtw, dtb,
                                                  xc, dtB, BmB, CmB);

    // selective scan + gate: 3072 lanes.
    scan_gate_kernel<<<(BATCH * D_INNER + 255) / 256, 256, 0, stream>>>(
        xz, xc, dtB, BmB, CmB, al, dp, yB);

    // h = y @ opw^T : 294 tile-waves, 4 waves/block -> 74 blocks (guarded).
    gemm_wmma_f32_nb4<<<74, GEMM_WPB * 32, 0, stream>>>(yB, opw, hid,
                                                        M_ROWS, D_MODEL, D_INNER);
  }

  // Final: res += h; out = LN(res).
  add_ln_kernel<<<196, 256, 0, stream>>>(hid, res, normf_w, normf_b, out);
}